// MambaBlock_29515015258140
// MI455X (gfx1250) — compile-verified
//
#include <hip/hip_runtime.h>
#include <math.h>

// Problem constants (match reference)
#define Hdim 256
#define Sdim 16
#define Bdim 2
#define Ndim 8192
#define Mrows (Bdim * Ndim)      // 16384 rows total
#define KDIM Hdim                // GEMM reduction dim = 256
#define NCH 64                   // scan chunks
#define CHL (Ndim / NCH)         // 128 steps per chunk
#define LN_EPS 1e-5f

typedef float v2f __attribute__((ext_vector_type(2)));
typedef float v8f __attribute__((ext_vector_type(8)));

__device__ __forceinline__ v2f ldv2(const float* p) {
  return *reinterpret_cast<const v2f*>(p);
}

// D = A(16x4,f32) * B(4x16,f32) + C(16x16,f32); probe-confirmed 8-arg pattern.
__device__ __forceinline__ v8f wmma4(v2f a, v2f b, v8f c) {
  return __builtin_amdgcn_wmma_f32_16x16x4_f32(false, a, false, b, (short)0, c,
                                               false, false);
}

__device__ __forceinline__ float softplusf(float x) {
  return x > 20.0f ? x : log1pf(expf(x));
}

// ---------------------------------------------------------------------------
// Kernel 1: xz[m][0..511] = x[m][:] . in_w[n][:] + in_b[n]
// One wave -> 16x64 output tile (4 accumulators, A fragment reused 4x).
// grid = 1024 blocks * 8 waves = 8192 waves = (16384/16) m-tiles * 8 n-tiles.
// ---------------------------------------------------------------------------
__global__ __launch_bounds__(256) void gemm_xz(const float* __restrict__ x,
                                               const float* __restrict__ in_w,
                                               const float* __restrict__ in_b,
                                               float* __restrict__ xz) {
  const int wave = (blockIdx.x * blockDim.x + threadIdx.x) >> 5;
  const int lane = threadIdx.x & 31;
  const int half = lane >> 4;
  const int l = lane & 15;
  const int tile_m = wave >> 3;   // 0..1023
  const int tile_n = wave & 7;    // 0..7
  const int m0 = tile_m * 16;
  const int n0 = tile_n * 64;

  v8f acc0 = {}, acc1 = {}, acc2 = {}, acc3 = {};
  // A fragment: lane<16 holds {A[l][k],A[l][k+1]}, lane>=16 holds {A[l][k+2],A[l][k+3]}
  const float* arow = x + (size_t)(m0 + l) * KDIM + 2 * half;
  const float* b0 = in_w + (size_t)(n0 + 0 * 16 + l) * KDIM + 2 * half;
  const float* b1 = in_w + (size_t)(n0 + 1 * 16 + l) * KDIM + 2 * half;
  const float* b2 = in_w + (size_t)(n0 + 2 * 16 + l) * KDIM + 2 * half;
  const float* b3 = in_w + (size_t)(n0 + 3 * 16 + l) * KDIM + 2 * half;

#pragma unroll 8
  for (int k = 0; k < KDIM; k += 4) {
    v2f a = ldv2(arow + k);
    acc0 = wmma4(a, ldv2(b0 + k), acc0);
    acc1 = wmma4(a, ldv2(b1 + k), acc1);
    acc2 = wmma4(a, ldv2(b2 + k), acc2);
    acc3 = wmma4(a, ldv2(b3 + k), acc3);
  }

  // C/D layout: vgpr r -> row (half*8 + r), col l within each 16-wide tile.
#pragma unroll
  for (int r = 0; r < 8; ++r) {
    const size_t row = (size_t)(m0 + half * 8 + r);
    const int c0 = n0 + l;
    xz[row * 512 + c0 + 0]  = acc0[r] + in_b[c0 + 0];
    xz[row * 512 + c0 + 16] = acc1[r] + in_b[c0 + 16];
    xz[row * 512 + c0 + 32] = acc2[r] + in_b[c0 + 32];
    xz[row * 512 + c0 + 48] = acc3[r] + in_b[c0 + 48];
  }
}

// ---------------------------------------------------------------------------
// Kernel 2: x_dbl[m][0..32] = x_inner[m][:] . xp_w[n][:] + xp_b[n]
// One wave -> 16x48 tile; columns >=33 clamped (loads valid, stores guarded),
// so no branches inside the WMMA loop and EXEC stays all-ones.
// grid = 128 blocks * 8 waves = 1024 m-tiles.
// ---------------------------------------------------------------------------
__global__ __launch_bounds__(256) void gemm_xdbl(const float* __restrict__ xz,
                                                 const float* __restrict__ xp_w,
                                                 const float* __restrict__ xp_b,
                                                 float* __restrict__ xdbl) {
  const int wave = (blockIdx.x * blockDim.x + threadIdx.x) >> 5;
  const int lane = threadIdx.x & 31;
  const int half = lane >> 4;
  const int l = lane & 15;
  const int m0 = wave * 16;

  v8f acc0 = {}, acc1 = {}, acc2 = {};
  const float* arow = xz + (size_t)(m0 + l) * 512 + 2 * half;  // x_inner, stride 512
  const int c1 = 16 + l;
  const int c2raw = 32 + l;
  const int c2 = c2raw < 33 ? c2raw : 32;  // clamp; garbage cols never stored
  const float* b0 = xp_w + (size_t)(l)*KDIM + 2 * half;
  const float* b1 = xp_w + (size_t)c1 * KDIM + 2 * half;
  const float* b2 = xp_w + (size_t)c2 * KDIM + 2 * half;

#pragma unroll 8
  for (int k = 0; k < KDIM; k += 4) {
    v2f a = ldv2(arow + k);
    acc0 = wmma4(a, ldv2(b0 + k), acc0);
    acc1 = wmma4(a, ldv2(b1 + k), acc1);
    acc2 = wmma4(a, ldv2(b2 + k), acc2);
  }

#pragma unroll
  for (int r = 0; r < 8; ++r) {
    const size_t row = (size_t)(m0 + half * 8 + r);
    xdbl[row * 33 + l] = acc0[r] + xp_b[l];
    xdbl[row * 33 + 16 + l] = acc1[r] + xp_b[16 + l];
    if (c2raw < 33) xdbl[row * 33 + c2raw] = acc2[r] + xp_b[c2raw];
  }
}

// ---------------------------------------------------------------------------
// Scan phase 1: per-(b,h,chunk) partial state sums (recurrence is a pure
// cumsum -- no decay -- so chunked prefix-sum is exact).
// 32768 threads: tid = b*16384 + chunk*256 + h  (h fastest for coalescing).
// ---------------------------------------------------------------------------
__global__ __launch_bounds__(256) void scan_partial(
    const float* __restrict__ xz, const float* __restrict__ xdbl,
    const float* __restrict__ dt_w, const float* __restrict__ dt_b,
    float* __restrict__ csum) {
  const int tid = blockIdx.x * 256 + threadIdx.x;
  const int h = tid & (Hdim - 1);
  const int c = (tid >> 8) & (NCH - 1);
  const int b = tid >> 14;
  const float dtw = dt_w[h];
  const float dtb = dt_b[h];

  float st[Sdim];
#pragma unroll
  for (int s = 0; s < Sdim; ++s) st[s] = 0.0f;

  const int n0 = c * CHL;
  for (int n = n0; n < n0 + CHL; ++n) {
    const size_t row = (size_t)(b * Ndim + n);
    const float xi = xz[row * 512 + h];
    const float* xd = xdbl + row * 33;
    const float dt = softplusf(xd[0] * dtw + dtb);
    const float u = xi * dt;
#pragma unroll
    for (int s = 0; s < Sdim; ++s) st[s] += u * xd[1 + s];
  }

  float* o = csum + ((size_t)(b * NCH + c) * Hdim + h) * Sdim;
#pragma unroll
  for (int s = 0; s < Sdim; ++s) o[s] = st[s];
}

// ---------------------------------------------------------------------------
// Scan phase 2: in-place exclusive prefix over the 64 chunk sums.
// 8192 threads: tid = b*4096 + h*16 + s (s fastest -> coalesced).
// ---------------------------------------------------------------------------
__global__ __launch_bounds__(256) void scan_prefix(float* __restrict__ csum) {
  const int tid = blockIdx.x * 256 + threadIdx.x;
  const int s = tid & (Sdim - 1);
  const int h = (tid >> 4) & (Hdim - 1);
  const int b = tid >> 12;
  float run = 0.0f;
  for (int c = 0; c < NCH; ++c) {
    const size_t idx = ((size_t)(b * NCH + c) * Hdim + h) * Sdim + s;
    const float t = csum[idx];
    csum[idx] = run;
    run += t;
  }
}

// ---------------------------------------------------------------------------
// Scan phase 3: replay chunk from its exclusive-prefix state; emit
// y = h.C + D*x_inner, gated by silu(z).
// ---------------------------------------------------------------------------
__global__ __launch_bounds__(256) void scan_apply(
    const float* __restrict__ xz, const float* __restrict__ xdbl,
    const float* __restrict__ dt_w, const float* __restrict__ dt_b,
    const float* __restrict__ Dv, const float* __restrict__ csum,
    float* __restrict__ y) {
  const int tid = blockIdx.x * 256 + threadIdx.x;
  const int h = tid & (Hdim - 1);
  const int c = (tid >> 8) & (NCH - 1);
  const int b = tid >> 14;
  const float dtw = dt_w[h];
  const float dtb = dt_b[h];
  const float dh = Dv[h];

  float st[Sdim];
  const float* pin = csum + ((size_t)(b * NCH + c) * Hdim + h) * Sdim;
#pragma unroll
  for (int s = 0; s < Sdim; ++s) st[s] = pin[s];

  const int n0 = c * CHL;
  for (int n = n0; n < n0 + CHL; ++n) {
    const size_t row = (size_t)(b * Ndim + n);
    const float xi = xz[row * 512 + h];
    const float* xd = xdbl + row * 33;
    const float dt = softplusf(xd[0] * dtw + dtb);
    const float u = xi * dt;
    float acc = dh * xi;
#pragma unroll
    for (int s = 0; s < Sdim; ++s) {
      st[s] += u * xd[1 + s];          // inclusive cumsum state
      acc += st[s] * xd[17 + s];       // dot with C_raw
    }
    const float zz = xz[row * 512 + Hdim + h];
    acc *= zz / (1.0f + expf(-zz));    // silu gate
    y[row * Hdim + h] = acc;
  }
}

// ---------------------------------------------------------------------------
// Kernel 6: out = y . out_w^T + out_b + residual, then LayerNorm over H=256.
// One block = 16 rows x 256 cols: 8 waves, each wave 2 WMMA accumulators.
// Tile staged in LDS, two-level reduction for mean/var.
// ---------------------------------------------------------------------------
__global__ __launch_bounds__(256) void gemm_out_ln(
    const float* __restrict__ y, const float* __restrict__ out_w,
    const float* __restrict__ out_b, const float* __restrict__ x,
    const float* __restrict__ ln_g, const float* __restrict__ ln_b,
    float* __restrict__ out) {
  __shared__ float tile[16][264];  // 264 stride breaks bank conflicts
  __shared__ float psum[16][16];
  __shared__ float psum2[16][16];
  __shared__ float stats[16][2];

  const int m0 = blockIdx.x * 16;
  const int w = threadIdx.x >> 5;   // wave 0..7
  const int lane = threadIdx.x & 31;
  const int half = lane >> 4;
  const int l = lane & 15;
  const int n0 = w * 32;

  v8f acc0 = {}, acc1 = {};
  const float* arow = y + (size_t)(m0 + l) * Hdim + 2 * half;
  const float* b0 = out_w + (size_t)(n0 + l) * Hdim + 2 * half;
  const float* b1 = out_w + (size_t)(n0 + 16 + l) * Hdim + 2 * half;

#pragma unroll 8
  for (int k = 0; k < KDIM; k += 4) {
    v2f a = ldv2(arow + k);
    acc0 = wmma4(a, ldv2(b0 + k), acc0);
    acc1 = wmma4(a, ldv2(b1 + k), acc1);
  }

#pragma unroll
  for (int r = 0; r < 8; ++r) {
    const int rowl = half * 8 + r;
    const size_t grow = (size_t)(m0 + rowl) * Hdim;
    const int c0 = n0 + l;
    tile[rowl][c0] = acc0[r] + out_b[c0] + x[grow + c0];
    tile[rowl][c0 + 16] = acc1[r] + out_b[c0 + 16] + x[grow + c0 + 16];
  }
  __syncthreads();

  // LayerNorm: 16 threads per row, each owns a 16-col segment.
  const int row = threadIdx.x >> 4;
  const int seg = threadIdx.x & 15;
  float s1 = 0.0f, s2 = 0.0f;
#pragma unroll
  for (int i = 0; i < 16; ++i) {
    const float v = tile[row][seg * 16 + i];
    s1 += v;
    s2 += v * v;
  }
  psum[row][seg] = s1;
  psum2[row][seg] = s2;
  __syncthreads();
  if (seg == 0) {
    float S = 0.0f, S2 = 0.0f;
#pragma unroll
    for (int i = 0; i < 16; ++i) {
      S += psum[row][i];
      S2 += psum2[row][i];
    }
    const float mu = S * (1.0f / 256.0f);
    const float var = S2 * (1.0f / 256.0f) - mu * mu;
    stats[row][0] = mu;
    stats[row][1] = rsqrtf(var + LN_EPS);
  }
  __syncthreads();
  const float mu = stats[row][0];
  const float rs = stats[row][1];
  const size_t grow = (size_t)(m0 + row) * Hdim;
#pragma unroll
  for (int i = 0; i < 16; ++i) {
    const int cidx = seg * 16 + i;
    const float v = tile[row][cidx];
    out[grow + cidx] = ln_g[cidx] * (v - mu) * rs + ln_b[cidx];
  }
}

// ---------------------------------------------------------------------------
extern "C" void kernel_launch(void* const* d_in, const int* in_sizes, int n_in,
                              void* d_out, int out_size, void* d_ws,
                              size_t ws_size, hipStream_t stream) {
  const float* x = (const float*)d_in[0];
  const float* in_w = (const float*)d_in[1];
  const float* in_b = (const float*)d_in[2];
  const float* xp_w = (const float*)d_in[3];
  const float* xp_b = (const float*)d_in[4];
  const float* dt_w = (const float*)d_in[5];
  const float* dt_b = (const float*)d_in[6];
  // d_in[7] = A_log: unused by the reference computation
  const float* Dv = (const float*)d_in[8];
  const float* out_w = (const float*)d_in[9];
  const float* out_b = (const float*)d_in[10];
  const float* ln_g = (const float*)d_in[11];
  const float* ln_bb = (const float*)d_in[12];
  float* out = (float*)d_out;

  float* ws = (float*)d_ws;
  float* xz = ws;                                  // 16384*512 = 8.4M f32
  float* xdbl = xz + (size_t)Mrows * 512;          // 16384*33
  float* csum = xdbl + (size_t)Mrows * 33;         // 2*64*256*16
  float* yb = csum + (size_t)Bdim * NCH * Hdim * Sdim;  // 16384*256

  gemm_xz<<<1024, 256, 0, stream>>>(x, in_w, in_b, xz);
  gemm_xdbl<<<128, 256, 0, stream>>>(xz, xp_w, xp_b, xdbl);
  scan_partial<<<128, 256, 0, stream>>>(xz, xdbl, dt_w, dt_b, csum);
  scan_prefix<<<32, 256, 0, stream>>>(csum);
  scan_apply<<<128, 256, 0, stream>>>(xz, xdbl, dt_w, dt_b, Dv, csum, yb);
  gemm_out_ln<<<1024, 256, 0, stream>>>(yb, out_w, out_b, x, ln_g, ln_bb, out);
}